// Cont_Loss_21930103014244
// MI455X (gfx1250) — compile-verified
//
#include <hip/hip_runtime.h>

// ---------------------------------------------------------------------------
// loss = sum((out[:, :, 0:16:2] - target[:, :, 1:16:2])^2) * 2^-22
// Shapes: (B=8, C=32, W=16, H=128, Wd=128) f32, row-major.
// Bandwidth-bound: 268 MB streamed once -> ~11.5 us floor @ 23.3 TB/s.
// Wave reduction uses V_WMMA_F32_16X16X4_F32 (B = ones => row sums of A).
// ---------------------------------------------------------------------------

typedef __attribute__((ext_vector_type(2))) float v2f;
typedef __attribute__((ext_vector_type(4))) float v4f;
typedef __attribute__((ext_vector_type(8))) float v8f;

#define ELEMS_TOTAL 33554432u          // 8*32*8*128*128 contributing elements
#define N4          (ELEMS_TOTAL / 4u) // float4 work items
#define NBLOCKS     2048
#define NTHREADS    256
#define NWAVES      (NTHREADS / 32)

__global__ void __launch_bounds__(NTHREADS)
cont_loss_partial(const float* __restrict__ outp,
                  const float* __restrict__ tgtp,
                  float* __restrict__ ws) {
    const unsigned tid    = blockIdx.x * NTHREADS + threadIdx.x;
    const unsigned stride = gridDim.x * NTHREADS;

    float acc = 0.0f;
    // N4 % (NBLOCKS*NTHREADS) == 0 -> uniform trip count, EXEC stays all-1s.
    for (unsigned i4 = tid; i4 < N4; i4 += stride) {
        const unsigned e   = i4 << 2;           // element index in even-W space
        const unsigned r   = e & 16383u;        // within H*Wd chunk
        const unsigned w2  = (e >> 14) & 7u;    // half-window slice
        const unsigned g   = e >> 17;           // b*32 + c
        const unsigned off = (g << 18) + (w2 << 15) + r;

        // Streamed once -> non-temporal b128 loads (don't thrash 192MB L2).
        const v4f o = __builtin_nontemporal_load(
            reinterpret_cast<const v4f*>(outp + off));
        const v4f t = __builtin_nontemporal_load(
            reinterpret_cast<const v4f*>(tgtp + off + 16384u));

        const float d0 = o.x - t.x, d1 = o.y - t.y;
        const float d2 = o.z - t.z, d3 = o.w - t.w;
        acc = fmaf(d0, d0, acc);
        acc = fmaf(d1, d1, acc);
        acc = fmaf(d2, d2, acc);
        acc = fmaf(d3, d3, acc);
    }

    // ---- wave32 reduction via V_WMMA_F32_16X16X4_F32 ----------------------
    // A layout (16x4 f32, 2 VGPRs): lanes 0-15 -> rows M=lane, VGPR0=K0;
    // lanes 16-31 -> rows M=lane-16, VGPR0=K2.  Put acc in VGPR0, 0 in VGPR1.
    // B = ones => D[m][n] = acc_m + acc_{m+16}.
    v2f a; a.x = acc;  a.y = 0.0f;
    v2f b; b.x = 1.0f; b.y = 1.0f;
    v8f c = {};
    c = __builtin_amdgcn_wmma_f32_16x16x4_f32(
            /*neg_a=*/false, a, /*neg_b=*/false, b,
            /*c_mod=*/(short)0, c, /*reuse_a=*/false, /*reuse_b=*/false);

    // D layout: VGPR v holds row M=v (lanes 0-15) / M=v+8 (lanes 16-31).
    // Per-lane sum of the 8 VGPRs = half the row sums; xor-16 completes it.
    float s = ((c[0] + c[1]) + (c[2] + c[3])) + ((c[4] + c[5]) + (c[6] + c[7]));
    s += __shfl_xor(s, 16, 32);

    __shared__ float lds[NWAVES];
    const unsigned lane = threadIdx.x & 31u;
    const unsigned wave = threadIdx.x >> 5;
    if (lane == 0) lds[wave] = s;
    __syncthreads();
    if (threadIdx.x == 0) {
        float bs = 0.0f;
        #pragma unroll
        for (int w = 0; w < NWAVES; ++w) bs += lds[w];
        ws[blockIdx.x] = bs;   // one f32 per block, fixed order
    }
}

__global__ void __launch_bounds__(NTHREADS)
cont_loss_final(const float* __restrict__ ws, float* __restrict__ out) {
    __shared__ float lds[NTHREADS];
    float s = 0.0f;
    for (unsigned i = threadIdx.x; i < NBLOCKS; i += NTHREADS) s += ws[i];
    lds[threadIdx.x] = s;
    __syncthreads();
    #pragma unroll
    for (int off = NTHREADS / 2; off > 0; off >>= 1) {
        if (threadIdx.x < (unsigned)off) lds[threadIdx.x] += lds[threadIdx.x + off];
        __syncthreads();
    }
    if (threadIdx.x == 0)
        out[0] = lds[0] * (1.0f / 4194304.0f);  // * 2^-22
}

extern "C" void kernel_launch(void* const* d_in, const int* in_sizes, int n_in,
                              void* d_out, int out_size, void* d_ws, size_t ws_size,
                              hipStream_t stream) {
    const float* outp = (const float*)d_in[0];   // 'out'    (8,32,16,128,128) f32
    const float* tgtp = (const float*)d_in[1];   // 'target' (8,32,16,128,128) f32
    float* ws   = (float*)d_ws;                  // NBLOCKS floats (8 KB)
    float* loss = (float*)d_out;                 // scalar f32

    cont_loss_partial<<<NBLOCKS, NTHREADS, 0, stream>>>(outp, tgtp, ws);
    cont_loss_final  <<<1,       NTHREADS, 0, stream>>>(ws, loss);
    (void)in_sizes; (void)n_in; (void)out_size; (void)ws_size;
}